// SAGE_36962488549638
// MI455X (gfx1250) — compile-verified
//
#include <hip/hip_runtime.h>

#define N_NODES 100000
#define N_EDGES 1600000
#define F_IN    64
#define K_TOT   128     // 2*F_IN
#define F_OUT   64
#define M_TILES (N_NODES / 16)   // 6250, exact
#define B_STRIDE 160    // floats per k-pair row in LDS (128 + 32 pad, bank-conflict-free)

typedef __attribute__((ext_vector_type(2))) float v2f;
typedef __attribute__((ext_vector_type(8))) float v8f;

// ---------------------------------------------------------------- zero ws
__global__ void SAGE_zero_kernel(float* __restrict__ p, int n) {
    int i = blockIdx.x * blockDim.x + threadIdx.x;
    int stride = gridDim.x * blockDim.x;
    for (; i < n; i += stride) p[i] = 0.0f;
}

// ------------------------------------------------- edge scatter (atomics)
// 16 lanes per edge; each lane moves 4 features with a float4 load and
// four global_atomic_add_f32 into the L2-resident accumulator.
__global__ __launch_bounds__(256) void SAGE_scatter_kernel(
    const float* __restrict__ h, const int* __restrict__ src,
    const int* __restrict__ dst, float* __restrict__ sum,
    float* __restrict__ deg) {
    int gid = blockIdx.x * 256 + threadIdx.x;
    int e = gid >> 4;
    if (e >= N_EDGES) return;
    int f0 = (gid & 15) << 2;           // feature group: 0,4,...,60
    int s = src[e];
    int d = dst[e];
    const float4 v = *(const float4*)(h + (size_t)s * F_IN + f0);
    float* o = sum + (size_t)d * F_IN + f0;
    atomicAdd(o + 0, v.x);
    atomicAdd(o + 1, v.y);
    atomicAdd(o + 2, v.z);
    atomicAdd(o + 3, v.w);
    if (f0 == 0) atomicAdd(deg + d, 1.0f);
}

// --------------------------------------------- mean normalize (in place)
__global__ __launch_bounds__(256) void SAGE_normalize_kernel(
    float* __restrict__ sum, const float* __restrict__ deg) {
    int i = blockIdx.x * 256 + threadIdx.x;
    if (i >= N_NODES * F_IN) return;
    float dg = deg[i >> 6];
    sum[i] *= 1.0f / fmaxf(dg, 1.0f);
}

// ------------------------------------- [h | h_N] @ W^T + b via f32 WMMA
// One wave per 16-row M-tile; 4 N-tiles of 16 (F_OUT=64), A-frag reused
// across the 4 v_wmma_f32_16x16x4_f32 issues per K-step; K=128 -> 32 steps.
// LDS holds W in K-pair-interleaved layout so each B fragment is one
// aligned ds_load_b64:  sB[(k>>1)*B_STRIDE + 2*n + (k&1)] = W[n*128 + k].
__global__ __launch_bounds__(256) void SAGE_gemm_wmma_kernel(
    const float* __restrict__ h, const float* __restrict__ hN,
    const float* __restrict__ W, const float* __restrict__ bias,
    float* __restrict__ out) {
    __shared__ float sB[(K_TOT / 2) * B_STRIDE];   // 40 KB
    for (int idx = threadIdx.x; idx < F_OUT * K_TOT; idx += 256) {
        int n = idx >> 7;        // output channel
        int k = idx & 127;       // k index
        sB[(k >> 1) * B_STRIDE + 2 * n + (k & 1)] = W[idx];
    }
    __syncthreads();

    int wave = threadIdx.x >> 5;
    int lane = threadIdx.x & 31;
    int tile = blockIdx.x * 8 + wave;
    if (tile >= M_TILES) return;

    int m0 = tile * 16;
    int ml = lane & 15;          // A row / B column within tile
    int hi = lane >> 4;          // selects K-pair {0,1} vs {2,3}
    int row = m0 + ml;

    // C/D layout: VGPR r, lane L -> D[M = r + 8*(L>>4)][N = L&15]
    // bias depends only on N -> same value in all 8 accumulator VGPRs.
    v8f acc[4];
    #pragma unroll
    for (int t = 0; t < 4; ++t) {
        float bv = bias[t * 16 + ml];
        #pragma unroll
        for (int r = 0; r < 8; ++r) acc[t][r] = bv;
    }

    const float* arow0 = h  + (size_t)row * F_IN;
    const float* arow1 = hN + (size_t)row * F_IN;
    // per-lane LDS base: k-pair row selected by hi, column pair by n
    const float* bbase = sB + (size_t)hi * B_STRIDE + 2 * ml;

    #pragma unroll
    for (int k0 = 0; k0 < K_TOT; k0 += 4) {
        int kk = k0 + hi * 2;
        v2f a;
        if (k0 < F_IN) {               // first 64 k's come from h
            a = *(const v2f*)(arow0 + kk);
        } else {                       // last 64 from mean-aggregated h_N
            a = *(const v2f*)(arow1 + (kk - F_IN));
        }
        #pragma unroll
        for (int t = 0; t < 4; ++t) {
            // (k0>>1)*B_STRIDE is a compile-time immediate after unroll
            v2f bm = *(const v2f*)(bbase + (k0 >> 1) * B_STRIDE + t * 32);
            acc[t] = __builtin_amdgcn_wmma_f32_16x16x4_f32(
                false, a, false, bm, (short)0, acc[t], false, false);
        }
    }

    #pragma unroll
    for (int t = 0; t < 4; ++t) {
        #pragma unroll
        for (int r = 0; r < 8; ++r) {
            out[(size_t)(m0 + hi * 8 + r) * F_OUT + t * 16 + ml] = acc[t][r];
        }
    }
}

// ---------------------------------------------------------------- launch
extern "C" void kernel_launch(void* const* d_in, const int* in_sizes, int n_in,
                              void* d_out, int out_size, void* d_ws, size_t ws_size,
                              hipStream_t stream) {
    const float* h    = (const float*)d_in[0];
    const int*   src  = (const int*)d_in[1];   // jax default x64-off => int32
    const int*   dst  = (const int*)d_in[2];
    const float* W    = (const float*)d_in[3];
    const float* bias = (const float*)d_in[4];
    float* out = (float*)d_out;

    float* sum = (float*)d_ws;                      // [N_NODES*F_IN]
    float* deg = sum + (size_t)N_NODES * F_IN;      // [N_NODES]

    // 1) zero accumulators (sum + deg are contiguous in ws)
    int nz = N_NODES * F_IN + N_NODES;
    SAGE_zero_kernel<<<1024, 256, 0, stream>>>(sum, nz);

    // 2) edge scatter: 16 lanes per edge
    long long sc_threads = (long long)N_EDGES * 16;
    int sc_blocks = (int)((sc_threads + 255) / 256);
    SAGE_scatter_kernel<<<sc_blocks, 256, 0, stream>>>(h, src, dst, sum, deg);

    // 3) mean normalize in place (sum becomes h_N)
    int nrm_blocks = (N_NODES * F_IN + 255) / 256;
    SAGE_normalize_kernel<<<nrm_blocks, 256, 0, stream>>>(sum, deg);

    // 4) fused concat + GEMM + bias via v_wmma_f32_16x16x4_f32
    int gemm_blocks = (M_TILES + 7) / 8;   // 8 waves (tiles) per block
    SAGE_gemm_wmma_kernel<<<gemm_blocks, 256, 0, stream>>>(h, sum, W, bias, out);
}